// Attn_7413113553589
// MI455X (gfx1250) — compile-verified
//
#include <hip/hip_runtime.h>
#include <hip/hip_bf16.h>

typedef __attribute__((ext_vector_type(2))) float v2f;
typedef __attribute__((ext_vector_type(8))) float v8f;

#define LL   512   // max_len
#define BB   32    // batch
#define H2   1024  // 2H
#define HH   512   // H (== E)
#define NFN  768   // feature embeddings
#define NTOT 1024  // NF + FP
#define RTOT (LL*BB)

// -------------------------------------------------------------------------
// Kernel 1: WH[r][n], r in [0,16384), n in [0,1024)
//   n <  512 : WH = enc . W_feat^T + b_feat   (wh2, E-dim)
//   n >= 512 : WH = enc . W_lin^T  + b_lin    (wh,  H-dim)
// -------------------------------------------------------------------------
#define K1M 64
#define K1N 128
#define K1K 32
#define PAD 36   // padded LDS row stride (floats): 144B = 16B-aligned, conflict-free

__global__ __launch_bounds__(256)
void gemm1_kernel(const float* __restrict__ enc,
                  const float* __restrict__ Wfeat, const float* __restrict__ bfeat,
                  const float* __restrict__ Wlin,  const float* __restrict__ blin,
                  float* __restrict__ WH)
{
    __shared__ float As[K1M][PAD];
    __shared__ float Bs[K1N][PAD];

    const int tid  = threadIdx.x;
    const int wave = tid >> 5;
    const int lane = tid & 31;
    const int row0 = blockIdx.x * K1M;
    const int col0 = blockIdx.y * K1N;
    const int wm   = (wave & 3) * 16;   // M-tile offset within block
    const int wn   = (wave >> 2) * 64;  // N-tile group offset within block
    const int m    = lane & 15;
    const int kk   = (lane >> 4) * 2;   // K sub-offset per WMMA operand layout

    v8f acc[4] = {};

    for (int k0 = 0; k0 < H2; k0 += K1K) {
        // Stage A: 64 rows x 32 k  (512 float4, 2 per thread)
        #pragma unroll
        for (int i = 0; i < 2; ++i) {
            int idx = tid + i * 256;
            int r = idx >> 3, c = (idx & 7) * 4;
            *(float4*)&As[r][c] =
                *(const float4*)&enc[(size_t)(row0 + r) * H2 + k0 + c];
        }
        // Stage B: 128 weight rows x 32 k  (1024 float4, 4 per thread)
        #pragma unroll
        for (int i = 0; i < 4; ++i) {
            int idx = tid + i * 256;
            int n = idx >> 3, c = (idx & 7) * 4;
            int gn = col0 + n;
            const float* wr = (gn < HH) ? (Wfeat + (size_t)gn * H2)
                                        : (Wlin  + (size_t)(gn - HH) * H2);
            *(float4*)&Bs[n][c] = *(const float4*)&wr[k0 + c];
        }
        __syncthreads();

        for (int k = 0; k < K1K; k += 4) {
            v2f a = *(const v2f*)&As[wm + m][k + kk];
            #pragma unroll
            for (int t = 0; t < 4; ++t) {
                v2f b = *(const v2f*)&Bs[wn + t * 16 + m][k + kk];
                acc[t] = __builtin_amdgcn_wmma_f32_16x16x4_f32(
                    false, a, false, b, (short)0, acc[t], false, false);
            }
        }
        __syncthreads();
    }

    // bias + store (lanes 0-15 cover M=wm..wm+7, lanes 16-31 cover M=wm+8..wm+15)
    const int hi8 = (lane >> 4) * 8;
    #pragma unroll
    for (int t = 0; t < 4; ++t) {
        int gn = col0 + wn + t * 16 + m;
        float bias = (gn < HH) ? bfeat[gn] : blin[gn - HH];
        #pragma unroll
        for (int j = 0; j < 8; ++j) {
            int gr = row0 + wm + hi8 + j;
            WH[(size_t)gr * NTOT + gn] = acc[t][j] + bias;
        }
    }
}

// -------------------------------------------------------------------------
// Kernel 2: per block: 16 rows (fixed b, l = l0..l0+15).
//   energy[i][n<768]  = WH[r_i][0:512]   . feat_emb[n]
//   energy[i][n>=768] = WH[r_i][512:1024]. v[n-768]
//   softmax over n, write out[b][n][l0+i] (contiguous along l via float4)
// -------------------------------------------------------------------------
#define LP 1028  // padded row stride: 16B-multiple, conflict-free A fetches

__global__ __launch_bounds__(512)
void gemm2_softmax_kernel(const float* __restrict__ WH,
                          const float* __restrict__ Femb,
                          const float* __restrict__ V,
                          float* __restrict__ out)
{
    __shared__ float tile[16 * LP];   // A during GEMM, then energy
    __shared__ float inv_sum[16];

    const int tid  = threadIdx.x;
    const int wave = tid >> 5;
    const int lane = tid & 31;
    const int l0   = blockIdx.x * 16;
    const int b    = blockIdx.y;
    const int m    = lane & 15;
    const int kk   = (lane >> 4) * 2;

    // Stage 16 WH rows (4 KB each, contiguous): 4096 float4, 8 per thread
    #pragma unroll
    for (int i = 0; i < 8; ++i) {
        int idx = tid + i * 512;
        int r = idx >> 8;             // 256 float4 per row
        int c = (idx & 255) * 4;
        *(float4*)&tile[r * LP + c] =
            *(const float4*)&WH[((size_t)(l0 + r) * BB + b) * NTOT + c];
    }
    __syncthreads();

    // Each wave: 4 column tiles, cols 64*wave + t*16 + m
    const int koff = (wave < 12) ? 0 : HH;        // wh2 half vs wh half of WH row
    const float* bp[4];
    #pragma unroll
    for (int t = 0; t < 4; ++t) {
        int n = wave * 64 + t * 16 + m;
        bp[t] = (wave < 12) ? (Femb + (size_t)n * HH)
                            : (V    + (size_t)(n - NFN) * HH);
    }

    v8f acc[4] = {};
    for (int k = 0; k < HH; k += 4) {
        v2f a = *(const v2f*)&tile[m * LP + koff + k + kk];
        #pragma unroll
        for (int t = 0; t < 4; ++t) {
            v2f bb = *(const v2f*)&bp[t][k + kk];
            acc[t] = __builtin_amdgcn_wmma_f32_16x16x4_f32(
                false, a, false, bb, (short)0, acc[t], false, false);
        }
    }
    __syncthreads();   // A no longer needed; reuse LDS for energy

    const int hi8 = (lane >> 4) * 8;
    #pragma unroll
    for (int t = 0; t < 4; ++t) {
        int n = wave * 64 + t * 16 + m;
        #pragma unroll
        for (int j = 0; j < 8; ++j)
            tile[(hi8 + j) * LP + n] = acc[t][j];
    }
    __syncthreads();

    // Softmax: wave i owns row i (1024 cols, 32 per lane)
    {
        float mx = -3.4e38f;
        for (int c = lane; c < NTOT; c += 32)
            mx = fmaxf(mx, tile[wave * LP + c]);
        #pragma unroll
        for (int off = 16; off > 0; off >>= 1)
            mx = fmaxf(mx, __shfl_xor(mx, off, 32));
        float s = 0.f;
        for (int c = lane; c < NTOT; c += 32) {
            float e = __expf(tile[wave * LP + c] - mx);
            tile[wave * LP + c] = e;
            s += e;
        }
        #pragma unroll
        for (int off = 16; off > 0; off >>= 1)
            s += __shfl_xor(s, off, 32);
        if (lane == 0) inv_sum[wave] = 1.0f / s;
    }
    __syncthreads();

    // Transposed write: out[b][n][l0+lq .. l0+lq+3] as float4 (contiguous in l)
    #pragma unroll
    for (int i = 0; i < 8; ++i) {
        int seg = tid + i * 512;        // 4096 segments: 1024 n x 4 l-quads
        int n  = seg >> 2;
        int lq = (seg & 3) * 4;
        float4 vv;
        vv.x = tile[(lq + 0) * LP + n] * inv_sum[lq + 0];
        vv.y = tile[(lq + 1) * LP + n] * inv_sum[lq + 1];
        vv.z = tile[(lq + 2) * LP + n] * inv_sum[lq + 2];
        vv.w = tile[(lq + 3) * LP + n] * inv_sum[lq + 3];
        *(float4*)&out[(size_t)b * NTOT * LL + (size_t)n * LL + l0 + lq] = vv;
    }
}

// -------------------------------------------------------------------------
extern "C" void kernel_launch(void* const* d_in, const int* in_sizes, int n_in,
                              void* d_out, int out_size, void* d_ws, size_t ws_size,
                              hipStream_t stream)
{
    const float* Femb  = (const float*)d_in[0];  // [768,512]
    const float* enc   = (const float*)d_in[1];  // [512,32,1024]
    const float* Wlin  = (const float*)d_in[2];  // [512,1024]
    const float* blin  = (const float*)d_in[3];  // [512]
    const float* Wfeat = (const float*)d_in[4];  // [512,1024]
    const float* bfeat = (const float*)d_in[5];  // [512]
    const float* V     = (const float*)d_in[6];  // [256,512]
    float* out = (float*)d_out;                  // [32,1024,512]
    float* WH  = (float*)d_ws;                   // 16384 x 1024 f32 = 64 MB

    gemm1_kernel<<<dim3(RTOT / K1M, NTOT / K1N), 256, 0, stream>>>(
        enc, Wfeat, bfeat, Wlin, blin, WH);
    gemm2_softmax_kernel<<<dim3(LL / 16, BB), 512, 0, stream>>>(
        WH, Femb, V, out);
}